// ROIPooling_38405597561618
// MI455X (gfx1250) — compile-verified
//
#include <hip/hip_runtime.h>
#include <hip/hip_bf16.h>

namespace {

constexpr int B_        = 32;
constexpr int H_        = 64;
constexpr int W_        = 64;
constexpr int C_        = 512;
constexpr int NBOX_     = 10;
constexpr int CROP_     = 7;
constexpr int STRIDEPX_ = 16;

// One block per (batch, iy, ix) cell. 128 threads = 4 wave32s; each thread
// moves one float4 (16B) of the 2KB channel vector via the gfx1250 async
// global->LDS->global data-mover path.
__global__ __launch_bounds__(128) void roi_pool_box0_async(
    const float* __restrict__ fmap,    // [B, H, W, C]
    const float* __restrict__ bboxes,  // [B, NBOX, 4]
    float* __restrict__ out)           // [B, CROP, CROP, C]
{
    __shared__ __align__(16) float tile[C_];  // 2KB staging, one cell

    const int cell = blockIdx.x % (CROP_ * CROP_);
    const int b    = blockIdx.x / (CROP_ * CROP_);
    const int iy   = cell / CROP_;
    const int ix   = cell % CROP_;
    const int tid  = threadIdx.x;

    // Box 0 of batch b: (cx, cy, bw, bh). Row pitch 160B -> 16B aligned.
    const float4 box = *reinterpret_cast<const float4*>(bboxes + (size_t)b * NBOX_ * 4);
    const float cx = box.x, cy = box.y, bw = box.z, bh = box.w;

    constexpr float inv_h = 1.0f / (float)(H_ * STRIDEPX_);
    constexpr float inv_w = 1.0f / (float)(W_ * STRIDEPX_);

    // Match reference op order exactly: (cy - bh/2) * inv, clipped to [0,1].
    const float y1 = fminf(fmaxf((cy - bh / 2.0f) * inv_h, 0.0f), 1.0f);
    const float x1 = fminf(fmaxf((cx - bw / 2.0f) * inv_w, 0.0f), 1.0f);
    const float y2 = fminf(fmaxf((cy + bh / 2.0f) * inv_h, 0.0f), 1.0f);
    const float x2 = fminf(fmaxf((cx + bw / 2.0f) * inv_w, 0.0f), 1.0f);

    // hs = (y2-y1)*(H-1)/(CROP-1), same association as reference.
    const float hs = (y2 - y1) * (float)(H_ - 1) / (float)(CROP_ - 1);
    const float ws = (x2 - x1) * (float)(W_ - 1) / (float)(CROP_ - 1);
    const float ys = y1 * (float)(H_ - 1) + (float)iy * hs;
    const float xs = x1 * (float)(W_ - 1) + (float)ix * ws;

    const bool in_bounds = (ys >= 0.0f) && (ys <= (float)(H_ - 1)) &&
                           (xs >= 0.0f) && (xs <= (float)(W_ - 1));

    int yi = (int)floorf(ys + 0.5f);
    int xi = (int)floorf(xs + 0.5f);
    yi = min(max(yi, 0), H_ - 1);
    xi = min(max(xi, 0), W_ - 1);

    float* dst = out + (((size_t)b * CROP_ + iy) * CROP_ + ix) * C_ + (size_t)tid * 4;

    if (in_bounds) {
        const float* src =
            fmap + (((size_t)b * H_ + yi) * W_ + xi) * C_ + (size_t)tid * 4;

        // Workgroup-relative LDS byte offset for this thread's 16B slot.
        const unsigned lds_off = (unsigned)(uintptr_t)(&tile[tid * 4]);

        // Async DMA: 16B/lane global -> LDS (512B per wave per instruction).
        asm volatile("global_load_async_to_lds_b128 %0, %1, off"
                     :
                     : "v"(lds_off), "v"(src)
                     : "memory");
        // Data must land in LDS before the async store reads it.
        asm volatile("s_wait_asynccnt 0x0" ::: "memory");
        // Async DMA: 16B/lane LDS -> global. s_endpgm's implicit wait-idle
        // guarantees completion before wave termination.
        asm volatile("global_store_async_from_lds_b128 %0, %1, off"
                     :
                     : "v"(dst), "v"(lds_off)
                     : "memory");
    } else {
        // Masked cell: reference emits zeros; d_out is poisoned, so write them.
        *reinterpret_cast<float4*>(dst) = make_float4(0.0f, 0.0f, 0.0f, 0.0f);
    }
}

}  // namespace

extern "C" void kernel_launch(void* const* d_in, const int* in_sizes, int n_in,
                              void* d_out, int out_size, void* d_ws, size_t ws_size,
                              hipStream_t stream) {
    (void)in_sizes; (void)n_in; (void)out_size; (void)d_ws; (void)ws_size;
    const float* fmap   = (const float*)d_in[0];  // [32,64,64,512] f32
    const float* bboxes = (const float*)d_in[1];  // [32,10,4] f32
    float* out          = (float*)d_out;          // [32,7,7,512] f32

    const dim3 grid(B_ * CROP_ * CROP_);  // 1568 blocks
    const dim3 block(128);                // 4 wave32s, one float4 per thread
    roi_pool_box0_async<<<grid, block, 0, stream>>>(fmap, bboxes, out);
}